// DIENModel_57999238365395
// MI455X (gfx1250) — compile-verified
//
#include <hip/hip_runtime.h>
#include <hip/hip_bf16.h>

#define B_SZ 1024
#define T_SZ 200
#define E_SZ 128
#define H_SZ 128
#define G3   384   // 3*H

typedef float v2f __attribute__((ext_vector_type(2)));
typedef float v8f __attribute__((ext_vector_type(8)));

// D = A(16x4 f32) * B(4x16 f32) + C(16x16 f32)
__device__ __forceinline__ v8f wmma4_f32(v2f a, v2f b, v8f c) {
  return __builtin_amdgcn_wmma_f32_16x16x4_f32(false, a, false, b, (short)0, c,
                                               false, false);
}

__device__ __forceinline__ float sigf(float x) { return 1.f / (1.f + __expf(-x)); }

// Async global->LDS copy of one dword per lane (CDNA5 path, ASYNCcnt-tracked).
// lds_off is a byte offset from the wave's LDS base; gsrc a per-lane address.
__device__ __forceinline__ void async_g2l_b32(unsigned lds_off, const float* gsrc) {
  asm volatile("global_load_async_to_lds_b32 %0, %1, off"
               :: "v"(lds_off), "v"(gsrc) : "memory");
}
__device__ __forceinline__ void async_wait0() {
  asm volatile("s_wait_asynccnt 0x0" ::: "memory");
}

// ---------------------------------------------------------------------------
// Kernel 1: xp[r, g] = sum_e item_emb[behavior_ids[r], e] * wih[g, e] + bih[g]
// r = b*T + t  (204800 rows), g in [0,384)
// grid = (12800, 6), block = 128 (4 waves, one 16x16 tile each)
// ---------------------------------------------------------------------------
__global__ void k_xp(const int* __restrict__ ids, const float* __restrict__ emb,
                     const float* __restrict__ wih, const float* __restrict__ bih,
                     float* __restrict__ xp) {
  const int wave = threadIdx.x >> 5, lane = threadIdx.x & 31;
  const int half = lane >> 4, l16 = lane & 15;
  const int r0 = blockIdx.x * 16;
  const int n0 = (blockIdx.y * 4 + wave) * 16;
  const float* arow = emb + (long)ids[r0 + l16] * E_SZ;
  const float* brow = wih + (long)(n0 + l16) * E_SZ;
  v8f acc = {};
  for (int k = 0; k < E_SZ; k += 4) {
    const int kk = k + 2 * half;
    v2f a, b;
    a.x = arow[kk]; a.y = arow[kk + 1];
    b.x = brow[kk]; b.y = brow[kk + 1];
    acc = wmma4_f32(a, b, acc);
  }
  const float bias = bih[n0 + l16];
  for (int i = 0; i < 8; ++i)
    xp[(long)(r0 + i + 8 * half) * G3 + n0 + l16] = acc[i] + bias;
}

// ---------------------------------------------------------------------------
// Kernel 2: GRU scan.  64 blocks x 256 threads; 16 batch rows per block.
// whh^T (128x384, 192KB) is DMA'd into LDS (async global->LDS, transposed
// placement so B-fragment reads are bank-conflict free) for all 200 steps.
// ---------------------------------------------------------------------------
__global__ void k_gru(const float* __restrict__ xp, const float* __restrict__ whh,
                      const float* __restrict__ bhh, float* __restrict__ hs) {
  extern __shared__ float lds[];
  float* whhT = lds;                 // [128][384]
  float* h    = whhT + 128 * G3;     // [16][128]
  float* hp   = h + 16 * H_SZ;       // [16][384]
  const int tid = threadIdx.x;
  const unsigned lds_base = (unsigned)__builtin_amdgcn_groupstaticsize();
  for (int i = tid; i < 128 * G3; i += 256) {
    const int k = i / G3, g = i % G3;
    async_g2l_b32(lds_base + (unsigned)i * 4u, &whh[g * H_SZ + k]);
  }
  for (int i = tid; i < 16 * H_SZ; i += 256) h[i] = 0.f;
  async_wait0();
  __syncthreads();

  const int b0 = blockIdx.x * 16;
  const int wave = tid >> 5, lane = tid & 31, half = lane >> 4, l16 = lane & 15;

  for (int t = 0; t < T_SZ; ++t) {
    // hp = h @ whh^T + bhh  (16 x 384); 24 n-tiles over 8 waves
    for (int j = 0; j < 3; ++j) {
      const int n0 = (wave * 3 + j) * 16;
      v8f acc = {};
      for (int k = 0; k < H_SZ; k += 4) {
        const int kk = k + 2 * half;
        v2f a, b;
        a.x = h[l16 * H_SZ + kk];       a.y = h[l16 * H_SZ + kk + 1];
        b.x = whhT[kk * G3 + n0 + l16]; b.y = whhT[(kk + 1) * G3 + n0 + l16];
        acc = wmma4_f32(a, b, acc);
      }
      const float bias = bhh[n0 + l16];
      for (int i = 0; i < 8; ++i)
        hp[(i + 8 * half) * G3 + n0 + l16] = acc[i] + bias;
    }
    __syncthreads();
    // gates (16x128 elements)
    for (int i = tid; i < 16 * H_SZ; i += 256) {
      const int m = i >> 7, c = i & 127;
      const long xr = (long)((b0 + m) * T_SZ + t) * G3;
      const float r = sigf(xp[xr + c]       + hp[m * G3 + c]);
      const float z = sigf(xp[xr + 128 + c] + hp[m * G3 + 128 + c]);
      const float n = tanhf(xp[xr + 256 + c] + r * hp[m * G3 + 256 + c]);
      const float hn = (1.f - z) * n + z * h[i];
      h[i] = hn;
      hs[(long)((b0 + m) * T_SZ + t) * H_SZ + c] = hn;
    }
    __syncthreads();
  }
}

// ---------------------------------------------------------------------------
// Kernel 3: aux[r] = relu(hs_row @ w1^T + b1) @ w2^T + b2, rows = b*199 + t
// 12736 blocks x 32 threads (one wave, 16 rows)
// ---------------------------------------------------------------------------
__global__ void k_aux(const float* __restrict__ hs, const float* __restrict__ w1,
                      const float* __restrict__ b1, const float* __restrict__ w2,
                      const float* __restrict__ b2, float* __restrict__ aux) {
  const int lane = threadIdx.x & 31, half = lane >> 4, l16 = lane & 15;
  const int r0 = blockIdx.x * 16;
  const int r = r0 + l16, b = r / 199, t = r % 199;
  const float* arow = hs + (long)(b * T_SZ + t) * H_SZ;
  float part[8] = {0, 0, 0, 0, 0, 0, 0, 0};
  for (int nt = 0; nt < 8; ++nt) {
    const int n0 = nt * 16;
    v8f acc = {};
    for (int k = 0; k < H_SZ; k += 4) {
      const int kk = k + 2 * half;
      v2f a, bf;
      a.x = arow[kk]; a.y = arow[kk + 1];
      bf.x = w1[(n0 + l16) * H_SZ + kk]; bf.y = w1[(n0 + l16) * H_SZ + kk + 1];
      acc = wmma4_f32(a, bf, acc);
    }
    const float bb = b1[n0 + l16], ww = w2[n0 + l16];
    for (int i = 0; i < 8; ++i) {
      float v = acc[i] + bb;
      part[i] += (v > 0.f ? v : 0.f) * ww;
    }
  }
  for (int off = 1; off < 16; off <<= 1)
    for (int i = 0; i < 8; ++i) part[i] += __shfl_xor(part[i], off, 32);
  if (l16 == 0)
    for (int i = 0; i < 8; ++i) aux[r0 + i + 8 * half] = part[i] + b2[0];
}

// ---------------------------------------------------------------------------
// Kernel 4: attention scores.  att_in = [hs, ce, hs*ce] (K=384), N=64, then
// prelu + dot(w2).  12800 blocks x 32 threads (16 rows per wave).
// ---------------------------------------------------------------------------
__global__ void k_att(const float* __restrict__ hs, const float* __restrict__ emb,
                      const int* __restrict__ cand_id, const float* __restrict__ w1,
                      const float* __restrict__ b1, const float* __restrict__ a_prm,
                      const float* __restrict__ w2, const float* __restrict__ b2,
                      float* __restrict__ scores) {
  const int lane = threadIdx.x & 31, half = lane >> 4, l16 = lane & 15;
  const int r0 = blockIdx.x * 16;
  const int r = r0 + l16, b = r / T_SZ;
  const float* hrow = hs + (long)r * H_SZ;
  const float* crow = emb + (long)cand_id[b] * E_SZ;
  float part[8] = {0, 0, 0, 0, 0, 0, 0, 0};
  const float alpha = a_prm[0];
  for (int nt = 0; nt < 4; ++nt) {
    const int n0 = nt * 16;
    v8f acc = {};
    for (int seg = 0; seg < 3; ++seg) {
      for (int k = 0; k < 128; k += 4) {
        const int kk = k + 2 * half;
        v2f a, bf;
        if (seg == 0)      { a.x = hrow[kk];            a.y = hrow[kk + 1]; }
        else if (seg == 1) { a.x = crow[kk];            a.y = crow[kk + 1]; }
        else               { a.x = hrow[kk] * crow[kk]; a.y = hrow[kk + 1] * crow[kk + 1]; }
        const int kc = seg * 128 + kk;
        bf.x = w1[(n0 + l16) * G3 + kc]; bf.y = w1[(n0 + l16) * G3 + kc + 1];
        acc = wmma4_f32(a, bf, acc);
      }
    }
    const float bb = b1[n0 + l16], ww = w2[n0 + l16];
    for (int i = 0; i < 8; ++i) {
      float v = acc[i] + bb;
      part[i] += (v >= 0.f ? v : alpha * v) * ww;
    }
  }
  for (int off = 1; off < 16; off <<= 1)
    for (int i = 0; i < 8; ++i) part[i] += __shfl_xor(part[i], off, 32);
  if (l16 == 0)
    for (int i = 0; i < 8; ++i) scores[r0 + i + 8 * half] = part[i] + b2[0];
}

// ---------------------------------------------------------------------------
// Kernel 5: softmax over T per batch row (in place). 1024 blocks x 256.
// ---------------------------------------------------------------------------
__global__ void k_softmax(float* __restrict__ s) {
  __shared__ float red[256];
  const int tid = threadIdx.x;
  float* row = s + (long)blockIdx.x * T_SZ;
  float m = -1e30f;
  for (int t = tid; t < T_SZ; t += 256) m = fmaxf(m, row[t]);
  red[tid] = m; __syncthreads();
  for (int st = 128; st > 0; st >>= 1) {
    if (tid < st) red[tid] = fmaxf(red[tid], red[tid + st]);
    __syncthreads();
  }
  m = red[0]; __syncthreads();
  float sum = 0.f;
  for (int t = tid; t < T_SZ; t += 256) sum += __expf(row[t] - m);
  red[tid] = sum; __syncthreads();
  for (int st = 128; st > 0; st >>= 1) {
    if (tid < st) red[tid] += red[tid + st];
    __syncthreads();
  }
  const float inv = 1.f / red[0];
  for (int t = tid; t < T_SZ; t += 256) row[t] = __expf(row[t] - m) * inv;
}

// ---------------------------------------------------------------------------
// Kernel 6: AUGRU time-independent gate halves:
//   xg[r, 0:128]   = hs_r @ wz[:, :H]^T + bz   (similarly wr, wh)
// grid = (12800, 6), block = 128
// ---------------------------------------------------------------------------
__global__ void k_xg(const float* __restrict__ hs, const float* __restrict__ wz,
                     const float* __restrict__ bz, const float* __restrict__ wr,
                     const float* __restrict__ br, const float* __restrict__ wh,
                     const float* __restrict__ bhg, float* __restrict__ xg) {
  const int wave = threadIdx.x >> 5, lane = threadIdx.x & 31;
  const int half = lane >> 4, l16 = lane & 15;
  const int nt = blockIdx.y * 4 + wave;      // 0..23
  const int gate = nt >> 3, n0 = (nt & 7) * 16;
  const float* w  = gate == 0 ? wz : (gate == 1 ? wr : wh);
  const float* bb = gate == 0 ? bz : (gate == 1 ? br : bhg);
  const int r0 = blockIdx.x * 16;
  const float* arow = hs + (long)(r0 + l16) * H_SZ;
  v8f acc = {};
  for (int k = 0; k < H_SZ; k += 4) {
    const int kk = k + 2 * half;
    v2f a, bf;
    a.x = arow[kk]; a.y = arow[kk + 1];
    bf.x = w[(n0 + l16) * 256 + kk]; bf.y = w[(n0 + l16) * 256 + kk + 1];
    acc = wmma4_f32(a, bf, acc);
  }
  const float bias = bb[n0 + l16];
  for (int i = 0; i < 8; ++i)
    xg[(long)(r0 + i + 8 * half) * G3 + gate * 128 + n0 + l16] = acc[i] + bias;
}

// ---------------------------------------------------------------------------
// Kernel 7: AUGRU scan. 64 blocks x 256 threads; h-gate weights (3x64KB)
// DMA'd into LDS via async global->LDS in transposed layout.
// ---------------------------------------------------------------------------
__global__ void k_augru(const float* __restrict__ xg, const float* __restrict__ att,
                        const float* __restrict__ wz, const float* __restrict__ wr,
                        const float* __restrict__ wh, float* __restrict__ hT) {
  extern __shared__ float lds[];
  float* WzT = lds;                  // [128][128] : wz[n*256 + 128 + k]
  float* WrT = WzT + 128 * 128;
  float* WhT = WrT + 128 * 128;
  float* h   = WhT + 128 * 128;      // [16][128]
  float* zs  = h + 16 * H_SZ;        // [16][128]
  float* rh  = zs + 16 * H_SZ;       // [16][128]
  const int tid = threadIdx.x;
  const unsigned lds_base = (unsigned)__builtin_amdgcn_groupstaticsize();
  for (int i = tid; i < 128 * 128; i += 256) {
    const int k = i >> 7, n = i & 127;
    const unsigned dst = lds_base + (unsigned)i * 4u;
    async_g2l_b32(dst,               &wz[n * 256 + 128 + k]);
    async_g2l_b32(dst + 65536u,      &wr[n * 256 + 128 + k]);
    async_g2l_b32(dst + 131072u,     &wh[n * 256 + 128 + k]);
  }
  for (int i = tid; i < 16 * H_SZ; i += 256) h[i] = 0.f;
  async_wait0();
  __syncthreads();

  const int b0 = blockIdx.x * 16;
  const int wave = tid >> 5, lane = tid & 31, half = lane >> 4, l16 = lane & 15;
  const int n0 = wave * 16;

  for (int t = 0; t < T_SZ; ++t) {
    v8f accz = {}, accr = {};
    for (int k = 0; k < H_SZ; k += 4) {
      const int kk = k + 2 * half;
      v2f a, bz_, br_;
      a.x = h[l16 * H_SZ + kk];         a.y = h[l16 * H_SZ + kk + 1];
      bz_.x = WzT[kk * 128 + n0 + l16]; bz_.y = WzT[(kk + 1) * 128 + n0 + l16];
      br_.x = WrT[kk * 128 + n0 + l16]; br_.y = WrT[(kk + 1) * 128 + n0 + l16];
      accz = wmma4_f32(a, bz_, accz);
      accr = wmma4_f32(a, br_, accr);
    }
    for (int i = 0; i < 8; ++i) {
      const int row = i + 8 * half, c = n0 + l16;
      const long xr = (long)((b0 + row) * T_SZ + t) * G3;
      const float z = sigf(xg[xr + c] + accz[i]);
      const float r = sigf(xg[xr + 128 + c] + accr[i]);
      zs[row * H_SZ + c] = z;
      rh[row * H_SZ + c] = r * h[row * H_SZ + c];
    }
    __syncthreads();
    v8f acch = {};
    for (int k = 0; k < H_SZ; k += 4) {
      const int kk = k + 2 * half;
      v2f a, bh_;
      a.x = rh[l16 * H_SZ + kk];        a.y = rh[l16 * H_SZ + kk + 1];
      bh_.x = WhT[kk * 128 + n0 + l16]; bh_.y = WhT[(kk + 1) * 128 + n0 + l16];
      acch = wmma4_f32(a, bh_, acch);
    }
    for (int i = 0; i < 8; ++i) {
      const int row = i + 8 * half, c = n0 + l16;
      const long xr = (long)((b0 + row) * T_SZ + t) * G3;
      const float htl = tanhf(xg[xr + 256 + c] + acch[i]);
      const float zp = att[(long)(b0 + row) * T_SZ + t] * zs[row * H_SZ + c];
      h[row * H_SZ + c] = (1.f - zp) * h[row * H_SZ + c] + zp * htl;
    }
    __syncthreads();
  }
  for (int i = tid; i < 16 * H_SZ; i += 256) hT[(long)b0 * H_SZ + i] = h[i];
}

// ---------------------------------------------------------------------------
// Kernel 8: fused final MLP (352->256->128->64, LN+PReLU each) + heads(3).
// 64 blocks x 256 threads; 16 batch rows per block.
// ---------------------------------------------------------------------------
__device__ void mlp_layer(const float* xin, int in_d, int out_d,
                          const float* __restrict__ w, const float* __restrict__ bias,
                          const float* __restrict__ g, const float* __restrict__ be,
                          const float* __restrict__ pa, float* ybuf, float* xnext,
                          int tid) {
  const int wave = tid >> 5, lane = tid & 31, half = lane >> 4, l16 = lane & 15;
  const int ntiles = out_d >> 4;
  for (int nt = wave; nt < ntiles; nt += 8) {
    const int n0 = nt * 16;
    v8f acc = {};
    for (int k = 0; k < in_d; k += 4) {
      const int kk = k + 2 * half;
      v2f a, bf;
      a.x = xin[l16 * in_d + kk];         a.y = xin[l16 * in_d + kk + 1];
      bf.x = w[(n0 + l16) * in_d + kk];   bf.y = w[(n0 + l16) * in_d + kk + 1];
      acc = wmma4_f32(a, bf, acc);
    }
    const float bb = bias[n0 + l16];
    for (int i = 0; i < 8; ++i)
      ybuf[(i + 8 * half) * out_d + n0 + l16] = acc[i] + bb;
  }
  __syncthreads();
  if (tid < 16) {
    const int m = tid;
    float mean = 0.f;
    for (int c = 0; c < out_d; ++c) mean += ybuf[m * out_d + c];
    mean /= (float)out_d;
    float var = 0.f;
    for (int c = 0; c < out_d; ++c) {
      const float d = ybuf[m * out_d + c] - mean;
      var += d * d;
    }
    var /= (float)out_d;
    const float inv = rsqrtf(var + 1e-5f);
    const float alpha = pa[0];
    for (int c = 0; c < out_d; ++c) {
      float v = (ybuf[m * out_d + c] - mean) * inv * g[c] + be[c];
      xnext[m * out_d + c] = (v >= 0.f ? v : alpha * v);
    }
  }
  __syncthreads();
}

__global__ void k_mlp(const float* __restrict__ hT, const float* __restrict__ emb,
                      const float* __restrict__ cemb, const int* __restrict__ cand_id,
                      const int* __restrict__ cand_cat, const float* __restrict__ dense,
                      const float* __restrict__ dp_w, const float* __restrict__ dp_b,
                      const float* w1, const float* b1, const float* g1, const float* be1, const float* pa1,
                      const float* w2, const float* b2, const float* g2, const float* be2, const float* pa2,
                      const float* w3, const float* b3, const float* g3, const float* be3, const float* pa3,
                      const float* __restrict__ hw, const float* __restrict__ hb,
                      float* __restrict__ preds) {
  __shared__ float xbuf[16 * 352];
  __shared__ float ybuf[16 * 256];
  const int tid = threadIdx.x;
  const int b0 = blockIdx.x * 16;
  for (int i = tid; i < 16 * 352; i += 256) {
    const int m = i / 352, c = i % 352, b = b0 + m;
    float v;
    if (c < 128)       v = hT[(long)b * H_SZ + c];
    else if (c < 256)  v = emb[(long)cand_id[b] * E_SZ + (c - 128)];
    else if (c < 320)  v = cemb[(long)cand_cat[b] * 64 + (c - 256)];
    else {
      const int j = c - 320;
      float s = dp_b[j];
      for (int q = 0; q < 5; ++q) s += dense[b * 5 + q] * dp_w[j * 5 + q];
      v = s;
    }
    xbuf[m * 352 + c] = v;
  }
  __syncthreads();
  mlp_layer(xbuf, 352, 256, w1, b1, g1, be1, pa1, ybuf, xbuf, tid);
  mlp_layer(xbuf, 256, 128, w2, b2, g2, be2, pa2, ybuf, xbuf, tid);
  mlp_layer(xbuf, 128,  64, w3, b3, g3, be3, pa3, ybuf, xbuf, tid);
  if (tid < 48) {
    const int m = tid / 3, j = tid % 3;
    float s = hb[j];
    for (int c = 0; c < 64; ++c) s += xbuf[m * 64 + c] * hw[j * 64 + c];
    preds[(long)(b0 + m) * 3 + j] = sigf(s);
  }
}

// ---------------------------------------------------------------------------
extern "C" void kernel_launch(void* const* d_in, const int* in_sizes, int n_in,
                              void* d_out, int out_size, void* d_ws, size_t ws_size,
                              hipStream_t stream) {
  const int*   behavior_ids = (const int*)  d_in[0];
  const int*   candidate_id = (const int*)  d_in[1];
  const int*   candidate_cat= (const int*)  d_in[2];
  const float* dense        = (const float*)d_in[3];
  const float* item_emb     = (const float*)d_in[4];
  const float* cat_emb      = (const float*)d_in[5];
  const float* gru_wih      = (const float*)d_in[6];
  const float* gru_whh      = (const float*)d_in[7];
  const float* gru_bih      = (const float*)d_in[8];
  const float* gru_bhh      = (const float*)d_in[9];
  const float* aux_w1       = (const float*)d_in[10];
  const float* aux_b1       = (const float*)d_in[11];
  const float* aux_w2       = (const float*)d_in[12];
  const float* aux_b2       = (const float*)d_in[13];
  const float* att_w1       = (const float*)d_in[14];
  const float* att_b1       = (const float*)d_in[15];
  const float* att_a        = (const float*)d_in[16];
  const float* att_w2       = (const float*)d_in[17];
  const float* att_b2       = (const float*)d_in[18];
  const float* wz           = (const float*)d_in[19];
  const float* bz           = (const float*)d_in[20];
  const float* wr           = (const float*)d_in[21];
  const float* br           = (const float*)d_in[22];
  const float* wh           = (const float*)d_in[23];
  const float* bh           = (const float*)d_in[24];
  const float* dp_w         = (const float*)d_in[25];
  const float* dp_b         = (const float*)d_in[26];
  const float* m_w1 = (const float*)d_in[27]; const float* m_b1 = (const float*)d_in[28];
  const float* ln_g1= (const float*)d_in[29]; const float* ln_b1= (const float*)d_in[30];
  const float* pa1  = (const float*)d_in[31];
  const float* m_w2 = (const float*)d_in[32]; const float* m_b2 = (const float*)d_in[33];
  const float* ln_g2= (const float*)d_in[34]; const float* ln_b2= (const float*)d_in[35];
  const float* pa2  = (const float*)d_in[36];
  const float* m_w3 = (const float*)d_in[37]; const float* m_b3 = (const float*)d_in[38];
  const float* ln_g3= (const float*)d_in[39]; const float* ln_b3= (const float*)d_in[40];
  const float* pa3  = (const float*)d_in[41];
  const float* heads_w = (const float*)d_in[42];
  const float* heads_b = (const float*)d_in[43];

  float* preds = (float*)d_out;                 // (B,3)
  float* aux   = preds + (size_t)B_SZ * 3;      // (B,199)

  float* ws  = (float*)d_ws;
  float* xp  = ws;                                        // B*T*384 (reused as xg)
  float* hs  = xp  + (size_t)B_SZ * T_SZ * G3;            // B*T*128
  float* att = hs  + (size_t)B_SZ * T_SZ * H_SZ;          // B*T
  float* hT  = att + (size_t)B_SZ * T_SZ;                 // B*128

  const size_t gru_lds   = (size_t)(128 * G3 + 16 * H_SZ + 16 * G3) * sizeof(float);
  const size_t augru_lds = (size_t)(3 * 128 * 128 + 3 * 16 * H_SZ) * sizeof(float);
  // Deterministic, idempotent opt-in to >64KB dynamic LDS (320KB/WGP on CDNA5).
  hipFuncSetAttribute((const void*)k_gru,
                      hipFuncAttributeMaxDynamicSharedMemorySize, (int)gru_lds);
  hipFuncSetAttribute((const void*)k_augru,
                      hipFuncAttributeMaxDynamicSharedMemorySize, (int)augru_lds);

  // 1. xp = gather(item_emb) @ wih^T + bih
  k_xp<<<dim3((B_SZ * T_SZ) / 16, 6), 128, 0, stream>>>(behavior_ids, item_emb,
                                                        gru_wih, gru_bih, xp);
  // 2. GRU scan -> hs
  k_gru<<<B_SZ / 16, 256, gru_lds, stream>>>(xp, gru_whh, gru_bhh, hs);
  // 3. aux head -> d_out aux region
  k_aux<<<(B_SZ * 199) / 16, 32, 0, stream>>>(hs, aux_w1, aux_b1, aux_w2, aux_b2, aux);
  // 4. attention scores
  k_att<<<(B_SZ * T_SZ) / 16, 32, 0, stream>>>(hs, item_emb, candidate_id, att_w1,
                                               att_b1, att_a, att_w2, att_b2, att);
  // 5. softmax over T (in place)
  k_softmax<<<B_SZ, 256, 0, stream>>>(att);
  // 6. AUGRU x-halves (reuse xp buffer)
  k_xg<<<dim3((B_SZ * T_SZ) / 16, 6), 128, 0, stream>>>(hs, wz, bz, wr, br, wh, bh, xp);
  // 7. AUGRU scan -> hT
  k_augru<<<B_SZ / 16, 256, augru_lds, stream>>>(xp, att, wz, wr, wh, hT);
  // 8. fused MLP + heads -> preds
  k_mlp<<<B_SZ / 16, 256, 0, stream>>>(hT, item_emb, cat_emb, candidate_id,
                                       candidate_cat, dense, dp_w, dp_b,
                                       m_w1, m_b1, ln_g1, ln_b1, pa1,
                                       m_w2, m_b2, ln_g2, ln_b2, pa2,
                                       m_w3, m_b3, ln_g3, ln_b3, pa3,
                                       heads_w, heads_b, preds);
}